// MultiEmbedding_84370337562892
// MI455X (gfx1250) — compile-verified
//
#include <hip/hip_runtime.h>

// ---- problem sizes (fixed by the reference) ----
#define N_  8
#define S_  2048
#define L_  8
#define K_  1024
#define D_  1024
#define NS_ (N_ * S_)          // 16384 output rows

typedef __attribute__((ext_vector_type(4))) unsigned int v4u;
typedef __attribute__((ext_vector_type(8))) int          v8i;
typedef __attribute__((ext_vector_type(4))) int          v4i;
typedef __attribute__((ext_vector_type(4))) float        v4f;

static constexpr int WAVES_PER_BLOCK = 4;
static constexpr int BLOCK_THREADS   = WAVES_PER_BLOCK * 32;
static constexpr int GRID_BLOCKS     = 512;
static constexpr int HALF_ROWS       = 4;                 // rows per TDM gather
static constexpr int HALF_ELEMS      = HALF_ROWS * D_;    // 4096 floats = 16 KB
static constexpr int HALF_BYTES      = HALF_ELEMS * 4;
static constexpr int LDS_ELEMS       = WAVES_PER_BLOCK * 2 * HALF_ELEMS;

// Issue one TDM gather of HALF_ROWS rows (16-bit indices, monotone) from the
// (L*K) x D fp32 tensor at gaddr into LDS byte offset lds_byte_off.
// Descriptor layout per CDNA5 ISA ch. 8 (D# groups 0..3), gather mode.
__device__ __forceinline__ void tdm_gather_rows4(unsigned lds_byte_off,
                                                 unsigned long long gaddr,
                                                 unsigned pk01, unsigned pk23) {
  v4u g0;
  g0.x = 0x80000001u;                              // gather_mode=1, idx16, count=1
  g0.y = lds_byte_off;                             // lds_addr [63:32]
  g0.z = (unsigned)gaddr;                          // global_addr [95:64]
  g0.w = (unsigned)(gaddr >> 32) | 0x80000000u;    // global_addr hi | type=2 at [127:126]

  v8i g1;
  g1[0] = 0x00020000;                  // data_size=2 (4B) at [17:16]; no pad/iterate/multicast
  g1[1] = (int)(D_ << 16);             // tensor_dim0 = 1024 (bits 63:48)
  g1[2] = (int)((L_ * K_) << 16);      // tensor_dim1 = 8192 (bits 95:80)
  g1[3] = (int)(D_ << 16);             // tile_dim0 = 1024 (bits 127:112)
  g1[4] = HALF_ROWS;                   // tile_dim1 = #valid indices = 4 (bits 143:128)
  g1[5] = D_;                          // tensor_dim0_stride = 1024 (bits 207:160)
  g1[6] = 0;
  g1[7] = 0;

  v4i g2; g2[0] = (int)pk01; g2[1] = (int)pk23; g2[2] = 0; g2[3] = 0;  // row_index_0..3
  v4i g3; g3[0] = 0; g3[1] = 0; g3[2] = 0; g3[3] = 0;

  v8i g23;
  g23[0] = g2[0]; g23[1] = g2[1]; g23[2] = g2[2]; g23[3] = g2[3];
  g23[4] = g3[0]; g23[5] = g3[1]; g23[6] = g3[2]; g23[7] = g3[3];

  __builtin_amdgcn_tensor_load_to_lds(g0, g1, g2, g3, g23, 0);
}

__global__ void __launch_bounds__(BLOCK_THREADS)
multi_embed_tdm_kernel(const unsigned* __restrict__ x_lo,   // int64 indices, low dwords
                       const float* __restrict__ w,
                       float* __restrict__ out) {
  // +4 pad floats that receive a real (volatile) store so LLVM cannot treat the
  // whole array as "never stored -> loads are undef".
  __shared__ __align__(16) float lds[LDS_ELEMS + 4];      // 128 KB + pad

  const int lane = threadIdx.x & 31;
  const int wave = threadIdx.x >> 5;
  const int wid  = blockIdx.x * WAVES_PER_BLOCK + wave;
  const int nw   = GRID_BLOCKS * WAVES_PER_BLOCK;

  // Real store to the shared array (pad word; value irrelevant, races benign).
  *(volatile float*)&lds[LDS_ELEMS] = 0.0f;

  float* buf0 = &lds[wave * 2 * HALF_ELEMS];
  float* buf1 = buf0 + HALF_ELEMS;
  const unsigned off0 = (unsigned)(wave * 2 * HALF_BYTES);  // static LDS starts at 0
  const unsigned off1 = off0 + HALF_BYTES;
  const unsigned long long gaddr = (unsigned long long)w;

  // Compute the 4 packed 16-bit row indices for each half of pair p.
  // Lanes 0..7 hold rowid = l*K + x[p,l]; readlane packs them into SGPRs.
  #define LOAD_PACKS(p, a01, a23, b01, b23)                                   \
    {                                                                         \
      const int l_ = lane & 7;                                                \
      unsigned xi_ = x_lo[(((size_t)(p)) * L_ + l_) * 2];                     \
      unsigned rid_ = (unsigned)(l_ * K_) + xi_;  /* < 8192, monotone in l */ \
      unsigned r0 = (unsigned)__builtin_amdgcn_readlane((int)rid_, 0);        \
      unsigned r1 = (unsigned)__builtin_amdgcn_readlane((int)rid_, 1);        \
      unsigned r2 = (unsigned)__builtin_amdgcn_readlane((int)rid_, 2);        \
      unsigned r3 = (unsigned)__builtin_amdgcn_readlane((int)rid_, 3);        \
      unsigned r4 = (unsigned)__builtin_amdgcn_readlane((int)rid_, 4);        \
      unsigned r5 = (unsigned)__builtin_amdgcn_readlane((int)rid_, 5);        \
      unsigned r6 = (unsigned)__builtin_amdgcn_readlane((int)rid_, 6);        \
      unsigned r7 = (unsigned)__builtin_amdgcn_readlane((int)rid_, 7);        \
      a01 = r0 | (r1 << 16); a23 = r2 | (r3 << 16);                           \
      b01 = r4 | (r5 << 16); b23 = r6 | (r7 << 16);                           \
    }

  int p = wid;
  if (p < NS_) {
    unsigned a01, a23, b01, b23;
    LOAD_PACKS(p, a01, a23, b01, b23);
    tdm_gather_rows4(off0, gaddr, a01, a23);   // rows l=0..3
    tdm_gather_rows4(off1, gaddr, b01, b23);   // rows l=4..7
  }

  for (; p < NS_; p += nw) {
    const int pn = p + nw;
    const bool more = pn < NS_;
    unsigned na01 = 0, na23 = 0, nb01 = 0, nb23 = 0;
    if (more) LOAD_PACKS(pn, na01, na23, nb01, nb23);

    v4f acc[8];

    // ---- buffer 0 (rows 0..3) ----
    // Wait for TDM; the escaped pointer operand + memory clobber tells the
    // compiler this asm may have written buf0, so the ds_loads below are live
    // and stay ordered after the wait.
    asm volatile("s_wait_tensorcnt 0x1" :: "v"(buf0) : "memory");
    #pragma unroll
    for (int c = 0; c < 8; ++c)
      acc[c] = *(const v4f*)&buf0[0 * D_ + c * 128 + 4 * lane];
    #pragma unroll
    for (int r = 1; r < HALF_ROWS; ++r)
      #pragma unroll
      for (int c = 0; c < 8; ++c)
        acc[c] += *(const v4f*)&buf0[r * D_ + c * 128 + 4 * lane];

    if (more) {
      asm volatile("s_wait_dscnt 0x0" :: "v"(buf0) : "memory");  // reads done before overwrite
      tdm_gather_rows4(off0, gaddr, na01, na23);
    }

    // ---- buffer 1 (rows 4..7) ----
    if (more) asm volatile("s_wait_tensorcnt 0x1" :: "v"(buf1) : "memory");
    else      asm volatile("s_wait_tensorcnt 0x0" :: "v"(buf1) : "memory");
    #pragma unroll
    for (int r = 0; r < HALF_ROWS; ++r)
      #pragma unroll
      for (int c = 0; c < 8; ++c)
        acc[c] += *(const v4f*)&buf1[r * D_ + c * 128 + 4 * lane];

    if (more) {
      asm volatile("s_wait_dscnt 0x0" :: "v"(buf1) : "memory");
      tdm_gather_rows4(off1, gaddr, nb01, nb23);
    }

    // ---- streaming store of the 4 KB output row ----
    v4f* op = (v4f*)(out + (size_t)p * D_);
    #pragma unroll
    for (int c = 0; c < 8; ++c)
      __builtin_nontemporal_store(acc[c], op + c * 32 + lane);
  }
  #undef LOAD_PACKS
}

extern "C" void kernel_launch(void* const* d_in, const int* in_sizes, int n_in,
                              void* d_out, int out_size, void* d_ws, size_t ws_size,
                              hipStream_t stream) {
  (void)in_sizes; (void)n_in; (void)out_size; (void)d_ws; (void)ws_size;
  const unsigned* x_lo = (const unsigned*)d_in[0];   // int64 indices; values < 1024 -> low dword
  const float*    w    = (const float*)d_in[1];
  float*          out  = (float*)d_out;
  multi_embed_tdm_kernel<<<GRID_BLOCKS, BLOCK_THREADS, 0, stream>>>(x_lo, w, out);
}